// MGCNLoss_52158082842660
// MI455X (gfx1250) — compile-verified
//
#include <hip/hip_runtime.h>

namespace {
constexpr int G  = 4096;
constexpr int C  = 10;
constexpr int NN = 8388608;
constexpr float EPSF = 1e-8f;

constexpr int THREADS   = 256;
constexpr int K1_BLOCKS = 256;   // segment-sum pass
constexpr int K2_BLOCKS = 256;   // KL pass (must be multiple of 64 for WMMA finalize)

// workspace layout (floats)
constexpr int WS_SUMP  = 0;          // [G]  per-graph sum of score_pos
constexpr int WS_SUMN  = G;          // [G]  per-graph sum of score_neg
constexpr int WS_CNT   = 2 * G;      // [G]  per-graph node counts
constexpr int WS_SCAL  = 3 * G;      // [0]=ce_sum [1]=mse_sum [2]=num_graphs
constexpr int WS_PART  = 3 * G + 64; // [K2_BLOCKS] per-block KL partials
constexpr int WS_TOTAL = WS_PART + K2_BLOCKS;
}

typedef float v2f __attribute__((ext_vector_type(2)));
typedef float v8f __attribute__((ext_vector_type(8)));

// ---------------------------------------------------------------- init
__global__ __launch_bounds__(THREADS)
void init_kernel(float* __restrict__ ws) {
    int i = blockIdx.x * blockDim.x + threadIdx.x;
    if (i < WS_TOTAL) ws[i] = 0.0f;
}

// ------------------------------------------- pass 1: LDS-privatized segment sums
__global__ __launch_bounds__(THREADS)
void segsum_kernel(const float* __restrict__ sp, const float* __restrict__ sn,
                   const int* __restrict__ batch, float* __restrict__ ws) {
    __shared__ float lp[G];
    __shared__ float ln_[G];
    __shared__ float lc[G];
    for (int i = threadIdx.x; i < G; i += THREADS) { lp[i] = 0.f; ln_[i] = 0.f; lc[i] = 0.f; }
    __syncthreads();

    constexpr int STRIDE = K1_BLOCKS * THREADS;
    constexpr int ITERS  = NN / STRIDE;          // 128, compile-time
    const int base = blockIdx.x * THREADS + threadIdx.x;
    #pragma unroll 4
    for (int it = 0; it < ITERS; ++it) {
        const int i = base + it * STRIDE;        // fully coalesced
        const int b = batch[i];
        const float p = sp[i];
        const float n = sn[i];
        atomicAdd(&lp[b], p);                    // ds_add_f32
        atomicAdd(&ln_[b], n);
        atomicAdd(&lc[b], 1.0f);
    }
    __syncthreads();

    float* gp = ws + WS_SUMP;
    float* gn = ws + WS_SUMN;
    float* gc = ws + WS_CNT;
    for (int i = threadIdx.x; i < G; i += THREADS) {
        atomicAdd(&gp[i], lp[i]);                // global_atomic_add_f32
        atomicAdd(&gn[i], ln_[i]);
        atomicAdd(&gc[i], lc[i]);
    }
}

// ------------------------------------------- pass 2: KL terms (L2-resident stream)
__global__ __launch_bounds__(THREADS)
void kl_kernel(const float* __restrict__ sp, const float* __restrict__ sn,
               const int* __restrict__ batch, float* __restrict__ ws) {
    __shared__ float invp[G];
    __shared__ float invn[G];
    __shared__ float wsum[THREADS / 32];

    const float* gp = ws + WS_SUMP;
    const float* gn = ws + WS_SUMN;
    for (int i = threadIdx.x; i < G; i += THREADS) {
        invp[i] = 1.0f / (gp[i] + EPSF);
        invn[i] = 1.0f / (gn[i] + EPSF);
    }
    __syncthreads();

    constexpr int STRIDE = K2_BLOCKS * THREADS;
    constexpr int ITERS  = NN / STRIDE;          // 128, compile-time
    const int base = blockIdx.x * THREADS + threadIdx.x;
    float acc = 0.0f;
    #pragma unroll 4
    for (int it = 0; it < ITERS; ++it) {
        const int i = base + it * STRIDE;
        const int b = batch[i];
        const float p = sp[i] * invp[b];
        const float n = sn[i] * invn[b];
        const float minv = 1.0f / (0.5f * (p + n) + EPSF);
        acc += p * __logf((p + EPSF) * minv) + n * __logf((n + EPSF) * minv);
    }

    // wave32 reduce, then cross-wave via LDS
    #pragma unroll
    for (int off = 16; off > 0; off >>= 1) acc += __shfl_down(acc, off, 32);
    if ((threadIdx.x & 31) == 0) wsum[threadIdx.x >> 5] = acc;
    __syncthreads();
    if (threadIdx.x == 0) {
        float s = 0.f;
        #pragma unroll
        for (int w = 0; w < THREADS / 32; ++w) s += wsum[w];
        ws[WS_PART + blockIdx.x] = s;            // one partial per block, no atomics
    }
}

// ------------------------------------------- CE + MSE + num_graphs (tiny)
__global__ __launch_bounds__(THREADS)
void ce_mse_kernel(const float* __restrict__ logits, const float* __restrict__ pp,
                   const float* __restrict__ pn, const int* __restrict__ targets,
                   float* __restrict__ ws) {
    const int g = blockIdx.x * blockDim.x + threadIdx.x;   // one graph per thread
    float ce = 0.f, mse = 0.f, ind = 0.f;
    if (g < G) {
        const float* l = logits + g * C;
        float mx = l[0];
        #pragma unroll
        for (int c = 1; c < C; ++c) mx = fmaxf(mx, l[c]);
        float se = 0.f;
        #pragma unroll
        for (int c = 0; c < C; ++c) se += __expf(l[c] - mx);
        const float lse = mx + __logf(se);
        ce = lse - l[targets[g]];
        const float* a = pp + g * C;
        const float* b = pn + g * C;
        #pragma unroll
        for (int c = 0; c < C; ++c) { const float d = a[c] + b[c] - 1.0f; mse += d * d; }
        ind = (ws[WS_CNT + g] > 0.0f) ? 1.0f : 0.0f;
    }
    __shared__ float sh[3][THREADS / 32];
    #pragma unroll
    for (int off = 16; off > 0; off >>= 1) {
        ce  += __shfl_down(ce,  off, 32);
        mse += __shfl_down(mse, off, 32);
        ind += __shfl_down(ind, off, 32);
    }
    const int w = threadIdx.x >> 5;
    if ((threadIdx.x & 31) == 0) { sh[0][w] = ce; sh[1][w] = mse; sh[2][w] = ind; }
    __syncthreads();
    if (threadIdx.x == 0) {
        float a = 0.f, b = 0.f, c2 = 0.f;
        #pragma unroll
        for (int i = 0; i < THREADS / 32; ++i) { a += sh[0][i]; b += sh[1][i]; c2 += sh[2][i]; }
        atomicAdd(&ws[WS_SCAL + 0], a);
        atomicAdd(&ws[WS_SCAL + 1], b);
        atomicAdd(&ws[WS_SCAL + 2], c2);
    }
}

// ------------------------------------------- finalize: WMMA f32 reduction of partials
// Sum K2_BLOCKS partials with V_WMMA_F32_16X16X4_F32: A = 64 data values (16x4),
// B = ones (4x16) -> D[m][n] = rowsum_m for all n. Column N=0 lives in lane 0
// (M=0..7, VGPR0..7) and lane 16 (M=8..15), per the documented C/D layout.
__global__ __launch_bounds__(32)
void finalize_kernel(const float* __restrict__ ws_ro, float* __restrict__ out) {
    const int lane = threadIdx.x;
    const int m    = lane & 15;
    const int half = lane >> 4;

    v8f acc = {};
    v2f bones;
    bones.x = 1.0f;
    bones.y = 1.0f;

    const float* partials = ws_ro + WS_PART;
    #pragma unroll
    for (int base = 0; base < K2_BLOCKS; base += 64) {
        // A layout (32-bit A, 16x4): lane l<16 holds A[l][0],A[l][1]; lane l+16 holds A[l][2],A[l][3]
        v2f a;
        a.x = partials[base + m * 4 + half * 2 + 0];
        a.y = partials[base + m * 4 + half * 2 + 1];
        acc = __builtin_amdgcn_wmma_f32_16x16x4_f32(
            /*neg_a=*/false, a, /*neg_b=*/false, bones,
            /*c_mod=*/(short)0, acc, /*reuse_a=*/false, /*reuse_b=*/false);
    }
    float s = acc[0] + acc[1] + acc[2] + acc[3] + acc[4] + acc[5] + acc[6] + acc[7];
    const float kl_total = __shfl(s, 0, 32) + __shfl(s, 16, 32);

    if (lane == 0) {
        const float ce_sum   = ws_ro[WS_SCAL + 0];
        const float mse_sum  = ws_ro[WS_SCAL + 1];
        const float numg     = ws_ro[WS_SCAL + 2];
        const float l_train  = ce_sum / (float)G;
        const float js_loss  = 0.5f * kl_total / numg;
        const float mse_loss = mse_sum / (float)(G * C);
        const float l_cor    = 1.0f * js_loss + 1.0f * mse_loss;   // ALPHA=BETA=1
        const float l_total  = l_train + 0.1f * l_cor;             // LAMBDA_COR=0.1
        out[0] = l_total;
        out[1] = l_train;
        out[2] = l_cor;
    }
}

// ---------------------------------------------------------------- launch
extern "C" void kernel_launch(void* const* d_in, const int* in_sizes, int n_in,
                              void* d_out, int out_size, void* d_ws, size_t ws_size,
                              hipStream_t stream) {
    const float* logits  = (const float*)d_in[0];  // [G,C]
    const float* pp      = (const float*)d_in[1];  // [G,C]
    const float* pn      = (const float*)d_in[2];  // [G,C]
    const float* sp      = (const float*)d_in[3];  // [NN]
    const float* sn      = (const float*)d_in[4];  // [NN]
    const int*   targets = (const int*)d_in[5];    // [G]
    const int*   batch   = (const int*)d_in[6];    // [NN]
    float* out = (float*)d_out;                    // 3 scalars
    float* ws  = (float*)d_ws;

    init_kernel<<<(WS_TOTAL + THREADS - 1) / THREADS, THREADS, 0, stream>>>(ws);
    segsum_kernel<<<K1_BLOCKS, THREADS, 0, stream>>>(sp, sn, batch, ws);
    kl_kernel<<<K2_BLOCKS, THREADS, 0, stream>>>(sp, sn, batch, ws);
    ce_mse_kernel<<<G / THREADS, THREADS, 0, stream>>>(logits, pp, pn, targets, ws);
    finalize_kernel<<<1, 32, 0, stream>>>(ws, out);
}